// CombinedModel_25563645346362
// MI455X (gfx1250) — compile-verified
//
#include <hip/hip_runtime.h>

#define DIM 128

typedef __attribute__((ext_vector_type(2))) float v2f;
typedef __attribute__((ext_vector_type(8))) float v8f;

// ---------------------------------------------------------------------------
// Zero-init output (d_out is poisoned with 0xAA by the harness).
// ---------------------------------------------------------------------------
__global__ void gnn_zero_f4(float4* __restrict__ p, int n4) {
    int i = blockIdx.x * blockDim.x + threadIdx.x;
    if (i < n4) p[i] = make_float4(0.f, 0.f, 0.f, 0.f);
}

// ---------------------------------------------------------------------------
// h = x @ W  via V_WMMA_F32_16X16X4_F32.
// Block = 256 threads = 8 waves. Block computes a 16-row strip of h;
// wave w computes the 16x16 tile at columns [16w, 16w+16).
// K is processed in chunks of 32 staged through LDS (padded vs bank conflicts).
// ---------------------------------------------------------------------------
__global__ __launch_bounds__(256) void gnn_gemm_wmma(
    const float* __restrict__ x, const float* __restrict__ W,
    float* __restrict__ h, int n_nodes) {
    __shared__ float sW[32][132];   // W[kc..kc+32) x [0..128), padded row
    __shared__ float sX[16][36];    // x[row0..row0+16) x [kc..kc+32), padded row

    const int t     = threadIdx.x;
    const int wave  = t >> 5;
    const int lane  = t & 31;
    const int m     = lane & 15;          // A: row in tile / B,D: col in tile
    const int khalf = (lane >> 4) << 1;   // 0 for lanes 0-15, 2 for lanes 16-31
    const int row0  = blockIdx.x * 16;
    const int col0  = wave * 16;

    v8f c = {0.f, 0.f, 0.f, 0.f, 0.f, 0.f, 0.f, 0.f};

    for (int kc = 0; kc < DIM; kc += 32) {
        // Stage W chunk: 32x128 floats = 1024 float4, 4 per thread.
        #pragma unroll
        for (int i = 0; i < 4; ++i) {
            int f   = t + 256 * i;        // float4 index within chunk
            int r   = f >> 5;             // 32 float4 per row
            int col = (f & 31) << 2;
            const float4 wv = *(const float4*)(W + (size_t)(kc + r) * DIM + col);
            sW[r][col + 0] = wv.x; sW[r][col + 1] = wv.y;
            sW[r][col + 2] = wv.z; sW[r][col + 3] = wv.w;
        }
        // Stage x chunk: 16x32 floats = 128 float4, threads 0-127.
        if (t < 128) {
            int r    = t >> 3;
            int col  = (t & 7) << 2;
            int grow = row0 + r;
            float4 xv = make_float4(0.f, 0.f, 0.f, 0.f);
            if (grow < n_nodes)
                xv = *(const float4*)(x + (size_t)grow * DIM + kc + col);
            sX[r][col + 0] = xv.x; sX[r][col + 1] = xv.y;
            sX[r][col + 2] = xv.z; sX[r][col + 3] = xv.w;
        }
        __syncthreads();

        #pragma unroll
        for (int k0 = 0; k0 < 32; k0 += 4) {
            v2f a, b;
            a.x = sX[m][k0 + khalf + 0];
            a.y = sX[m][k0 + khalf + 1];
            b.x = sW[k0 + khalf + 0][col0 + m];
            b.y = sW[k0 + khalf + 1][col0 + m];
            // 8 args: (neg_a, A, neg_b, B, c_mod, C, reuse_a, reuse_b)
            c = __builtin_amdgcn_wmma_f32_16x16x4_f32(
                    false, a, false, b, (short)0, c, false, false);
        }
        __syncthreads();
    }

    // D layout: VGPR i -> row (i + 8*(lane>=16)), col (lane%16).
    const int mbase = (lane >> 4) << 3;
    #pragma unroll
    for (int i = 0; i < 8; ++i) {
        int grow = row0 + mbase + i;
        if (grow < n_nodes)
            h[(size_t)grow * DIM + col0 + m] = c[i];
    }
}

// ---------------------------------------------------------------------------
// Edge scatter: one wave (32 lanes) per edge; each lane moves 4 floats.
// out[dst] += h[src] with hardware fp32 atomics (dst array is L2-resident).
// ---------------------------------------------------------------------------
__global__ __launch_bounds__(256) void gnn_edge_scatter(
    const float* __restrict__ h, const int* __restrict__ edge,
    float* __restrict__ out, int n_edges) {
    int gw   = (blockIdx.x * 256 + threadIdx.x) >> 5;   // global wave = edge id
    int lane = threadIdx.x & 31;
    if (gw >= n_edges) return;

    int s = edge[gw];             // edge_index[0][e] (source)
    int d = edge[n_edges + gw];   // edge_index[1][e] (destination)

    const float4 v = *(const float4*)(h + (size_t)s * DIM + (lane << 2));
    float* o = out + (size_t)d * DIM + (lane << 2);
    unsafeAtomicAdd(o + 0, v.x);
    unsafeAtomicAdd(o + 1, v.y);
    unsafeAtomicAdd(o + 2, v.z);
    unsafeAtomicAdd(o + 3, v.w);
}

// ---------------------------------------------------------------------------
// In-place relu on the accumulated output.
// ---------------------------------------------------------------------------
__global__ void gnn_relu_f4(float4* __restrict__ p, int n4) {
    int i = blockIdx.x * blockDim.x + threadIdx.x;
    if (i < n4) {
        float4 v = p[i];
        p[i] = make_float4(fmaxf(v.x, 0.f), fmaxf(v.y, 0.f),
                           fmaxf(v.z, 0.f), fmaxf(v.w, 0.f));
    }
}

extern "C" void kernel_launch(void* const* d_in, const int* in_sizes, int n_in,
                              void* d_out, int out_size, void* d_ws, size_t ws_size,
                              hipStream_t stream) {
    const float* x    = (const float*)d_in[0];
    const int*   edge = (const int*)d_in[1];   // [2, E] flattened, row-major
    const float* W    = (const float*)d_in[2];
    float*       out  = (float*)d_out;
    float*       h    = (float*)d_ws;          // [n_nodes, 128] scratch (25.6 MB)

    const int n_nodes = in_sizes[0] / DIM;
    const int n_edges = in_sizes[1] / 2;
    const int n_out   = n_nodes * DIM;
    const int n4      = n_out / 4;

    gnn_zero_f4<<<(n4 + 255) / 256, 256, 0, stream>>>((float4*)out, n4);
    gnn_gemm_wmma<<<(n_nodes + 15) / 16, 256, 0, stream>>>(x, W, h, n_nodes);
    gnn_edge_scatter<<<(n_edges + 7) / 8, 256, 0, stream>>>(h, edge, out, n_edges);
    gnn_relu_f4<<<(n4 + 255) / 256, 256, 0, stream>>>((float4*)out, n4);
}